// EmbeddingLayer_5394478924093
// MI455X (gfx1250) — compile-verified
//
#include <hip/hip_runtime.h>

typedef __attribute__((ext_vector_type(2))) float v2f;
typedef __attribute__((ext_vector_type(4))) float v4f;
typedef __attribute__((ext_vector_type(8))) float v8f;

constexpr int FDIM = 64;          // feature dim
constexpr int HDIM = 64;          // hidden dim
constexpr int BT   = 32 * 512;    // flattened batch*time
constexpr int TTILE = 16;         // bt rows per block (WMMA M)
constexpr int HS = 68;            // LDS h-stride (floats): pad 64->68 for banks, 16B aligned
constexpr int TS = 1092;          // LDS t-stride (floats): >=16*HS, 16B aligned
constexpr int LDS_FLOATS = TTILE * TS;   // 17472 floats = 69888 bytes

// out[bt, h, f] = in[bt, f] * w[f, h] + bias[f, h]
// Per f: rank-1 outer product over (t,h) => V_WMMA_F32_16X16X4_F32 with K=1 live.
__global__ void __launch_bounds__(256)
embedding_wmma_kernel(const float* __restrict__ inp,   // [BT, F]
                      const float* __restrict__ wgt,   // [F, H]
                      const float* __restrict__ bia,   // [F, H]
                      float* __restrict__ out) {       // [BT, H, F]
    extern __shared__ float lds[];

    const int tid  = threadIdx.x;
    const int lane = tid & 31;
    const int wave = tid >> 5;               // 0..7, owns f = 8*wave .. 8*wave+7
    const int bt0  = blockIdx.x * TTILE;
    const bool lo  = lane < 16;
    const int ln16 = lane & 15;
    const int f0   = wave * 8;

    // Hoist A operand: A vgpr0 lanes0-15 = in[bt0+lane, f] (K=0); everything else 0.
    // Lanes 16-31 carry K=2 in vgpr0 -> must be zero. vgpr1 (K=1,3) zero.
    const float* inRow = inp + (size_t)(bt0 + (lo ? lane : 0)) * FDIM;
    float av[8];
    #pragma unroll
    for (int fi = 0; fi < 8; ++fi)
        av[fi] = lo ? inRow[f0 + fi] : 0.0f;

    for (int ht = 0; ht < 4; ++ht) {
        const int h0 = ht * 16;

        #pragma unroll
        for (int fi = 0; fi < 8; ++fi) {
            const int f = f0 + fi;

            v2f a; a.x = av[fi]; a.y = 0.0f;

            // B vgpr0 lanes0-15 = row K=0: w[f, h0+n]; all other K rows zero.
            v2f b; b.x = lo ? wgt[f * HDIM + h0 + ln16] : 0.0f; b.y = 0.0f;

            // C[t,h] = bias[f,h] for all t -> splat bias[f, h0 + (lane&15)] in 8 vgprs.
            const float bv = bia[f * HDIM + h0 + ln16];
            v8f c = {bv, bv, bv, bv, bv, bv, bv, bv};

            v8f d = __builtin_amdgcn_wmma_f32_16x16x4_f32(
                false, a, false, b, (short)0, c, false, false);

            // D layout: vgpr r -> t = r + 8*(lane>=16), h = lane&15.
            // Scatter into padded LDS slab [t][h][f].
            const int tTop = (lane >> 4) * 8;
            float* p = lds + ln16 * HS + f;
            #pragma unroll
            for (int r = 0; r < 8; ++r)
                p[(tTop + r) * TS] = d[r];
        }
        __syncthreads();

        // Coalesced write-out: out[(bt0+t), h0+h, f], 16h x 64f contiguous per t.
        // 256 threads: h = tid>>4 (0..15), f = 4*(tid&15). Wave stores 512B contiguous.
        const int h  = tid >> 4;
        const int fq = (tid & 15) * 4;
        const float* lp = lds + h * HS + fq;
        float* gp = out + (size_t)bt0 * (HDIM * FDIM) + (h0 + h) * FDIM + fq;
        #pragma unroll
        for (int t = 0; t < TTILE; ++t) {
            v4f v = *(const v4f*)(lp + t * TS);
            // Non-temporal: 268MB output stream, don't rinse L2.
            __builtin_nontemporal_store(v, (v4f*)(gp + (size_t)t * (HDIM * FDIM)));
        }
        __syncthreads();
    }
}

extern "C" void kernel_launch(void* const* d_in, const int* in_sizes, int n_in,
                              void* d_out, int out_size, void* d_ws, size_t ws_size,
                              hipStream_t stream) {
    const float* inp = (const float*)d_in[0];   // (32,512,64) f32
    const float* wgt = (const float*)d_in[1];   // (64,64) f32
    const float* bia = (const float*)d_in[2];   // (64,64) f32
    float* out = (float*)d_out;                 // (32,512,64,64) f32

    dim3 grid(BT / TTILE);   // 1024 blocks
    dim3 block(256);         // 8 waves
    size_t shmem = LDS_FLOATS * sizeof(float);  // 69888 B dynamic LDS
    hipLaunchKernelGGL(embedding_wmma_kernel, grid, block, shmem, stream,
                       inp, wgt, bia, out);
}